// EntropyLoss_53008486367390
// MI455X (gfx1250) — compile-verified
//
#include <hip/hip_runtime.h>
#include <math.h>

// Problem constants (from reference): L=8, B=8, C=512, H=W=64
#define L_DIM   8
#define B_DIM   8
#define C_DIM   512
#define D_DIM   4096            // H*W
#define NSAMP   (L_DIM * B_DIM) // 64 stacked [C,H,W] samples
#define KSEL    (C_DIM / 10)    // 51 -> 52nd smallest (index k of ascending sort)
#define NBLK    (C_DIM / 64)    // 8 row-blocks of 64
#define NPAIR   (NBLK * (NBLK + 1) / 2) // 36 upper-triangular 64x64 tile pairs

typedef __attribute__((ext_vector_type(2))) float v2f;
typedef __attribute__((ext_vector_type(8))) float v8f;

// -----------------------------------------------------------------------------
// Kernel 1: Gram matrices G[s] = F_s * F_s^T via V_WMMA_F32_16X16X4_F32.
// One wave computes one 64x64 tile (16 wmma accumulators, K-loop over 4096).
// A fragment (16x4 f32): lane l holds F[rA + (l&15)][k0 + 2*(l>>4) + {0,1}].
// B fragment ("column-major" 4x16) of F^T has the identical per-lane layout,
// so both operands use the same float2 load. Symmetry: mirror-store bi != bj.
// -----------------------------------------------------------------------------
__global__ __launch_bounds__(256) void gram_kernel(const float* __restrict__ in,
                                                   float* __restrict__ gram) {
    const int lane = threadIdx.x & 31;
    const int wave = threadIdx.x >> 5;
    const int job  = blockIdx.x * 8 + wave;      // 0 .. NSAMP*NPAIR-1 (2304)

    const int s = job / NPAIR;
    int p = job % NPAIR;
    int bi = 0;
    while (p >= NBLK - bi) { p -= NBLK - bi; ++bi; }  // wave-uniform
    const int bj = bi + p;                            // bi <= bj
    const int rA = bi * 64;
    const int rB = bj * 64;

    const size_t fbase = (size_t)s * C_DIM * D_DIM;
    const int mrow = lane & 15;           // M index within 16-row fragment
    const int koff = (lane >> 4) * 2;     // lanes 16-31 hold K = k0+2, k0+3

    const float* Fa[4];
    const float* Fb[4];
#pragma unroll
    for (int i = 0; i < 4; ++i) {
        Fa[i] = in + fbase + (size_t)(rA + i * 16 + mrow) * D_DIM + koff;
        Fb[i] = in + fbase + (size_t)(rB + i * 16 + mrow) * D_DIM + koff;
    }

    v8f acc[4][4] = {};                   // 16 fp32 16x16 accumulators

    for (int k0 = 0; k0 < D_DIM; k0 += 4) {
        v2f a[4], b[4];
#pragma unroll
        for (int i = 0; i < 4; ++i) a[i] = *(const v2f*)(Fa[i] + k0);
#pragma unroll
        for (int j = 0; j < 4; ++j) b[j] = *(const v2f*)(Fb[j] + k0);
#pragma unroll
        for (int i = 0; i < 4; ++i)
#pragma unroll
            for (int j = 0; j < 4; ++j)
                acc[i][j] = __builtin_amdgcn_wmma_f32_16x16x4_f32(
                    /*neg_a=*/false, a[i], /*neg_b=*/false, b[j],
                    /*c_mod=*/(short)0, acc[i][j],
                    /*reuse_a=*/false, /*reuse_b=*/false);
    }

    // C/D layout: VGPR r, lanes 0-15 -> M=r, lanes 16-31 -> M=r+8; N = lane&15
    const int chi = (lane >> 4) * 8;
    float* gs = gram + (size_t)s * C_DIM * C_DIM;
#pragma unroll
    for (int i = 0; i < 4; ++i)
#pragma unroll
        for (int j = 0; j < 4; ++j)
#pragma unroll
            for (int r = 0; r < 8; ++r) {
                const int row = rA + i * 16 + r + chi;
                const int col = rB + j * 16 + (lane & 15);
                const float v = acc[i][j][r];
                gs[(size_t)row * C_DIM + col] = v;
                if (bi != bj) gs[(size_t)col * C_DIM + row] = v;
            }
}

// -----------------------------------------------------------------------------
// Kernel 2: per-row distances + exact k-th smallest (stable rank select).
// One 512-thread block per (sample, row). diag(G) provides the squared norms.
// -----------------------------------------------------------------------------
__global__ __launch_bounds__(512) void kth_kernel(const float* __restrict__ gram,
                                                  float* __restrict__ kth) {
    __shared__ float dist[C_DIM];
    const int s = blockIdx.x >> 9;            // blockIdx = s*512 + n
    const int n = blockIdx.x & (C_DIM - 1);
    const int t = threadIdx.x;

    const float* gs = gram + (size_t)s * C_DIM * C_DIM;
    const float dnn = gs[(size_t)n * C_DIM + n];
    const float dtt = gs[(size_t)t * C_DIM + t];
    const float gnt = gs[(size_t)n * C_DIM + t];
    const float d2  = dnn + dtt - 2.0f * gnt;
    const float d   = sqrtf(fmaxf(d2, 1e-12f));   // matches clip(.,1e-12)+sqrt
    dist[t] = d;
    __syncthreads();

    int cnt = 0;
    for (int m = 0; m < C_DIM; ++m) {
        const float dv = dist[m];                 // LDS broadcast read
        cnt += (dv < d) || (dv == d && m < t);    // stable rank; ranks are a
    }                                             // permutation of 0..511
    if (cnt == KSEL) kth[blockIdx.x] = d;         // exactly one writer per row
}

// -----------------------------------------------------------------------------
// Kernel 3: deterministic per-layer reduction -> entropies -> variance loss.
// -----------------------------------------------------------------------------
__global__ __launch_bounds__(256) void final_kernel(const float* __restrict__ kth,
                                                    float* __restrict__ out) {
    __shared__ float red[256];
    __shared__ float ents[L_DIM];
    const int t = threadIdx.x;

    for (int l = 0; l < L_DIM; ++l) {
        float p = 0.0f;
        for (int i = t; i < B_DIM * C_DIM; i += 256)
            p += kth[l * (B_DIM * C_DIM) + i];
        red[t] = p;
        __syncthreads();
        for (int off = 128; off > 0; off >>= 1) {
            if (t < off) red[t] += red[t + off];
            __syncthreads();
        }
        if (t == 0) ents[l] = logf(red[0] + 1.0f);
        __syncthreads();
    }

    if (t == 0) {
        // h = L/2 - 1 = 3
        const float d1a = ents[2] - ents[1];
        const float d1b = ents[3] - ents[2];
        const float m1  = 0.5f * (d1a + d1b);
        const float v1  = (d1a - m1) * (d1a - m1) + (d1b - m1) * (d1b - m1); // /(2-1)

        float dd[4];
        dd[0] = ents[4] - ents[3];
        dd[1] = ents[5] - ents[4];
        dd[2] = ents[6] - ents[5];
        dd[3] = ents[7] - ents[6];
        const float m2 = 0.25f * (dd[0] + dd[1] + dd[2] + dd[3]);
        float v2 = 0.0f;
        for (int i = 0; i < 4; ++i) v2 += (dd[i] - m2) * (dd[i] - m2);
        v2 *= (1.0f / 3.0f);                                                 // ddof=1

        out[0] = v1 + v2;   // LOSS_WEIGHT = 1.0
    }
}

// -----------------------------------------------------------------------------
// Workspace layout (requires ws_size >= 64 MB + 128 KB):
//   [0, 64MB)          : gram, 64 samples x 512 x 512 f32
//   [64MB, 64MB+128KB) : kth,  64 samples x 512 f32
// -----------------------------------------------------------------------------
extern "C" void kernel_launch(void* const* d_in, const int* in_sizes, int n_in,
                              void* d_out, int out_size, void* d_ws, size_t ws_size,
                              hipStream_t stream) {
    (void)in_sizes; (void)n_in; (void)out_size; (void)ws_size;
    const float* in = (const float*)d_in[0];
    float* gram = (float*)d_ws;
    float* kth  = (float*)((char*)d_ws +
                           (size_t)NSAMP * C_DIM * C_DIM * sizeof(float));

    // 2304 wave-jobs (64 samples x 36 tile pairs), 8 waves per 256-thread block
    gram_kernel<<<(NSAMP * NPAIR) / 8, 256, 0, stream>>>(in, gram);
    kth_kernel<<<NSAMP * C_DIM, 512, 0, stream>>>(gram, kth);
    final_kernel<<<1, 256, 0, stream>>>(kth, (float*)d_out);
}